// GNN_51539607552522
// MI455X (gfx1250) — compile-verified
//
#include <hip/hip_runtime.h>
#include <math.h>

// ---------------------------------------------------------------------------
// Problem constants (from the reference): C_IN=256, HID=10, B=8, H=W=120, N_P=6
// ---------------------------------------------------------------------------
#define NPIX_IMG 14400            // H*W
#define NB 8
#define HID 10
#define NPIX (NB * NPIX_IMG)      // 115200 pixels, divisible by 16 and 256
#define TILES (NPIX / 16)         // 7200 WMMA tiles

typedef float v2f __attribute__((ext_vector_type(2)));
typedef float v8f __attribute__((ext_vector_type(8)));

// Param indices in d_in[5 + i], following setup_inputs() insertion order.
enum {
  comp_f_W_ = 0, comp_f_g_, comp_f_b_,
  upd_f_Wz_, upd_f_bz_, upd_f_Wr_, upd_f_br_, upd_f_Wn_, upd_f_bn_,
  upd_u_Wz_, upd_u_bz_, upd_u_Wr_, upd_u_br_, upd_u_Wn_, upd_u_bn_,
  upd_l_Wz_, upd_l_bz_, upd_l_Wr_, upd_l_br_, upd_l_Wn_, upd_l_bn_,
  pdp_h_W_, pdp_h_g_, pdp_h_b_,
  att_h_W_, att_h_b_,
  comp_phu_W_, comp_phu_g_, comp_phu_b_,
  comp_phl_W_, comp_phl_g_, comp_phl_b_,
  dec_fhu_Wa_, dec_fhu_ba_, dec_fhu_Wm_, dec_fhu_g_, dec_fhu_b_,
  dec_fhl_Wa_, dec_fhl_ba_, dec_fhl_Wm_, dec_fhl_g_, dec_fhl_b_,
  pdp_p_W_, pdp_p_g_, pdp_p_b_,
  att_p_W_, att_p_b_,
  dec_hp_Wa_, dec_hp_ba_, dec_hp_Wm_, dec_hp_g_, dec_hp_b_,
  upd_p_Wz_, upd_p_bz_, upd_p_Wr_, upd_p_br_, upd_p_Wn_, upd_p_bn_,
  NW_
};

// ---------------------------------------------------------------------------
// Kernel A: fp32 WMMA projection of the C_IN=256 feature maps.
//   h-side outputs (per pixel): n=0..9  fea_t      = pdp_h_W[:, :256] . h_fea
//                               n=10    dec_fhu_Wa[0,:256] . h_fea
//                               n=11    dec_fhl_Wa[0,:256] . h_fea
//   p-side outputs (per pixel): n=0..9  fea_tp     = pdp_p_W[:, :256] . p_fea
//                               n=10..15 dec_hp_Wa[p,0,:256] . p_fea
// One wave = one 16-pixel tile; K=256 swept with V_WMMA_F32_16X16X4_F32.
// A layout (ISA 7.12.2): lane l -> M=l%16, VGPR0=K(2*(l/16)), VGPR1=K+1.
// B layout assumed mirror: VGPR0 holds rows {K, K+2}, VGPR1 rows {K+1, K+3}.
// D layout (ISA table): lane l, VGPR r -> M=r+8*(l/16), N=l%16.
// ---------------------------------------------------------------------------
__global__ __launch_bounds__(256) void proj_wmma(
    const float* __restrict__ pfea, const float* __restrict__ hfea,
    const float* __restrict__ pdp_h_W, const float* __restrict__ fhuWa,
    const float* __restrict__ fhlWa, const float* __restrict__ pdp_p_W,
    const float* __restrict__ hpWa,
    float* __restrict__ hproj, float* __restrict__ pproj)
{
  // B panels in LDS, K-pair interleaved: entry e=(kp*16+n) holds (B[2kp][n], B[2kp+1][n])
  __shared__ float2 sBh[2048];
  __shared__ float2 sBp[2048];
  const int tid = threadIdx.x;
  for (int e = tid; e < 2048; e += 256) {
    const int kp = e >> 4, n = e & 15, k0 = kp * 2;
    float h0 = 0.f, h1 = 0.f, p0 = 0.f, p1 = 0.f;
    if (n < 10) {
      h0 = pdp_h_W[n * 266 + k0];  h1 = pdp_h_W[n * 266 + k0 + 1];
      p0 = pdp_p_W[n * 266 + k0];  p1 = pdp_p_W[n * 266 + k0 + 1];
    } else {
      if (n == 10) { h0 = fhuWa[k0]; h1 = fhuWa[k0 + 1]; }
      if (n == 11) { h0 = fhlWa[k0]; h1 = fhlWa[k0 + 1]; }
      p0 = hpWa[(n - 10) * 276 + k0];
      p1 = hpWa[(n - 10) * 276 + k0 + 1];
    }
    sBh[e] = make_float2(h0, h1);
    sBp[e] = make_float2(p0, p1);
  }
  __syncthreads();

  const int wid = blockIdx.x * 8 + (tid >> 5);
  if (wid >= TILES) return;
  const int lane = tid & 31;
  const int mcol = lane & 15;       // A: pixel row M ; D: output column N
  const int kh   = lane >> 4;       // K half-select

  const int t    = wid;
  const int bimg = (t * 16) / NPIX_IMG;
  const int rem0 = (t * 16) % NPIX_IMG;
  const float* Ah = hfea + (size_t)bimg * 256 * NPIX_IMG + rem0 + mcol;
  const float* Ap = pfea + (size_t)bimg * 256 * NPIX_IMG + rem0 + mcol;

  v8f ch = {};
  v8f cp = {};
  #pragma unroll 4
  for (int kb = 0; kb < 256; kb += 4) {
    const int ka = kb + 2 * kh;
    v2f ah, ap, bh, bp;
    ah.x = Ah[(size_t)ka * NPIX_IMG];
    ah.y = Ah[(size_t)(ka + 1) * NPIX_IMG];
    ap.x = Ap[(size_t)ka * NPIX_IMG];
    ap.y = Ap[(size_t)(ka + 1) * NPIX_IMG];
    const float2 wh = sBh[((kb >> 1) + kh) * 16 + mcol];
    const float2 wp = sBp[((kb >> 1) + kh) * 16 + mcol];
    bh.x = wh.x; bh.y = wh.y;
    bp.x = wp.x; bp.y = wp.y;
    ch = __builtin_amdgcn_wmma_f32_16x16x4_f32(false, ah, false, bh, (short)0, ch, false, false);
    cp = __builtin_amdgcn_wmma_f32_16x16x4_f32(false, ap, false, bp, (short)0, cp, false, false);
  }

  float* ho = hproj + (size_t)t * 256;   // [pixel-in-tile][16 outputs]
  float* po = pproj + (size_t)t * 256;
  #pragma unroll
  for (int r = 0; r < 8; r++) {
    const int mm = r + 8 * kh;
    ho[mm * 16 + mcol] = ch[r];
    po[mm * 16 + mcol] = cp[r];
  }
}

// ---------------------------------------------------------------------------
// Kernel B: fully fused per-pixel GNN step (all HID=10 matvecs + gating +
// 10-edge segment-max + 9 GRUs), everything in registers, weights via
// wave-uniform scalar loads.
// ---------------------------------------------------------------------------
struct KArgs {
  const float* xp; const float* xh; const float* xf;
  const float* hproj; const float* pproj;
  float* out;
  const float* w[NW_];
};

__device__ __forceinline__ float sigm(float x)  { return 1.f / (1.f + __expf(-x)); }
__device__ __forceinline__ float reluf(float x) { return fmaxf(x, 0.f); }

__device__ __forceinline__ void gru10(const float* x, const float* mm,
    const float* __restrict__ Wz, const float* __restrict__ bz,
    const float* __restrict__ Wr, const float* __restrict__ br,
    const float* __restrict__ Wn, const float* __restrict__ bn,
    float* outv)
{
  float z[HID], r[HID];
  #pragma unroll
  for (int o = 0; o < HID; o++) {
    float sz = bz[o], sr = br[o];
    #pragma unroll
    for (int c = 0; c < HID; c++) {
      sz = fmaf(Wz[o * 20 + c], x[c], sz);
      sr = fmaf(Wr[o * 20 + c], x[c], sr);
      sz = fmaf(Wz[o * 20 + 10 + c], mm[c], sz);
      sr = fmaf(Wr[o * 20 + 10 + c], mm[c], sr);
    }
    z[o] = sigm(sz); r[o] = sigm(sr);
  }
  #pragma unroll
  for (int o = 0; o < HID; o++) {
    float sn = bn[o];
    #pragma unroll
    for (int c = 0; c < HID; c++) {
      sn = fmaf(Wn[o * 20 + c], r[c] * x[c], sn);
      sn = fmaf(Wn[o * 20 + 10 + c], mm[c], sn);
    }
    const float n = tanhf(sn);
    outv[o] = (1.f - z[o]) * x[o] + z[o] * n;
  }
}

__global__ __launch_bounds__(256) void fused_pw(KArgs a)
{
  const int pix = blockIdx.x * 256 + threadIdx.x;
  if (pix >= NPIX) return;
  const int bimg = pix / NPIX_IMG;
  const int rem  = pix - bimg * NPIX_IMG;
  const size_t po = (size_t)bimg * HID * NPIX_IMG + rem;   // channel base for this pixel

  // Flat output offsets (return-tuple order).
  const size_t OXP = 0;               // xp_new  [6,8,10,hw]
  const size_t OXH = 6912000;         // xh_new  [2,8,10,hw]
  const size_t OXF = 9216000;         // xf_new  [8,10,hw]
  const size_t OHA = 10368000;        // hp_att  [6,8,1,hw]
  const size_t OAT = 11059200;        // att u/l [2,8,1,hw]
  const size_t OMF = 11289600;        // msg_f   [8,10,hw]

  // ---- load state ----
  float vf[HID], vu[HID], vl[HID];
  #pragma unroll
  for (int c = 0; c < HID; c++) {
    vf[c] = a.xf[po + (size_t)c * NPIX_IMG];
    vu[c] = a.xh[po + (size_t)c * NPIX_IMG];
    vl[c] = a.xh[(size_t)NB * HID * NPIX_IMG + po + (size_t)c * NPIX_IMG];
  }
  float vp[6][HID];
  #pragma unroll
  for (int p = 0; p < 6; p++)
    #pragma unroll
    for (int c = 0; c < HID; c++)
      vp[p][c] = a.xp[(size_t)p * NB * HID * NPIX_IMG + po + (size_t)c * NPIX_IMG];

  float hq[12], pq[16];
  #pragma unroll
  for (int n = 0; n < 12; n++) hq[n] = a.hproj[(size_t)pix * 16 + n];
  #pragma unroll
  for (int n = 0; n < 16; n++) pq[n] = a.pproj[(size_t)pix * 16 + n];

  float* out = a.out;

  // ---- msg_f = bnrelu(comp_f_W . [xf,xu,xl]) ----
  float msg[HID];
  {
    const float* W = a.w[comp_f_W_]; const float* g = a.w[comp_f_g_]; const float* b = a.w[comp_f_b_];
    #pragma unroll
    for (int o = 0; o < HID; o++) {
      float s = 0.f;
      #pragma unroll
      for (int c = 0; c < HID; c++) {
        s = fmaf(W[o * 30 + c],      vf[c], s);
        s = fmaf(W[o * 30 + 10 + c], vu[c], s);
        s = fmaf(W[o * 30 + 20 + c], vl[c], s);
      }
      msg[o] = reluf(s * g[o] + b[o]);
      out[OMF + po + (size_t)o * NPIX_IMG] = msg[o];
    }
  }
  // ---- xf_new = GRU(xf, msg_f) ----
  {
    float xfn[HID];
    gru10(vf, msg, a.w[upd_f_Wz_], a.w[upd_f_bz_], a.w[upd_f_Wr_], a.w[upd_f_br_],
          a.w[upd_f_Wn_], a.w[upd_f_bn_], xfn);
    #pragma unroll
    for (int o = 0; o < HID; o++) out[OXF + po + (size_t)o * NPIX_IMG] = xfn[o];
  }
  // ---- dp_u / dp_l (pdp_h with precomputed fea_t = hq[0..9]) ----
  float dpu[HID], dpl[HID];
  {
    const float* W = a.w[pdp_h_W_]; const float* g = a.w[pdp_h_g_]; const float* b = a.w[pdp_h_b_];
    #pragma unroll
    for (int o = 0; o < HID; o++) {
      float su = hq[o], sl = hq[o];
      #pragma unroll
      for (int c = 0; c < HID; c++) {
        su = fmaf(W[o * 266 + 256 + c], vu[c], su);
        sl = fmaf(W[o * 266 + 256 + c], vl[c], sl);
      }
      dpu[o] = reluf(su * g[o] + b[o]);
      dpl[o] = reluf(sl * g[o] + b[o]);
    }
  }
  // ---- gate = att_h[u] * att_h[l] ----
  float gate;
  {
    float su = a.w[att_h_b_][0], sl = a.w[att_h_b_][1];
    #pragma unroll
    for (int c = 0; c < HID; c++) {
      su = fmaf(a.w[att_h_W_][c],      vu[c], su);
      sl = fmaf(a.w[att_h_W_][10 + c], vl[c], sl);
    }
    gate = sigm(su) * sigm(sl);
  }
  // ---- upper branch: comp_phu + xlh + decomp(fhu) -> GRU(upd_u) ----
  {
    float mm[HID];
    const float* W = a.w[comp_phu_W_]; const float* g = a.w[comp_phu_g_]; const float* b = a.w[comp_phu_b_];
    #pragma unroll
    for (int o = 0; o < HID; o++) {
      float s = 0.f;
      #pragma unroll
      for (int c = 0; c < HID; c++) {
        s = fmaf(W[o * 50 + c],      vu[c],    s);
        s = fmaf(W[o * 50 + 10 + c], vp[0][c], s);
        s = fmaf(W[o * 50 + 20 + c], vp[1][c], s);
        s = fmaf(W[o * 50 + 30 + c], vp[2][c], s);
        s = fmaf(W[o * 50 + 40 + c], vp[3][c], s);
      }
      mm[o] = reluf(s * g[o] + b[o]);
    }
    float au = hq[10] + a.w[dec_fhu_ba_][0];
    const float* Wa = a.w[dec_fhu_Wa_];
    #pragma unroll
    for (int c = 0; c < HID; c++) {
      au = fmaf(Wa[256 + c], vf[c], au);
      au = fmaf(Wa[266 + c], vu[c], au);
    }
    au = sigm(au);
    out[OAT + (size_t)bimg * NPIX_IMG + rem] = au;
    const float* Wm = a.w[dec_fhu_Wm_]; const float* gg = a.w[dec_fhu_g_]; const float* bb = a.w[dec_fhu_b_];
    #pragma unroll
    for (int o = 0; o < HID; o++) {
      float s = 0.f;
      #pragma unroll
      for (int c = 0; c < HID; c++) s = fmaf(Wm[o * 10 + c], vf[c], s);
      const float xfhu = au * reluf(s * gg[o] + bb[o]);
      mm[o] += gate * dpl[o] + (1.f - gate) * vu[o] + xfhu;   // + xlh + xfhu
    }
    float xun[HID];
    gru10(vu, mm, a.w[upd_u_Wz_], a.w[upd_u_bz_], a.w[upd_u_Wr_], a.w[upd_u_br_],
          a.w[upd_u_Wn_], a.w[upd_u_bn_], xun);
    #pragma unroll
    for (int o = 0; o < HID; o++) out[OXH + po + (size_t)o * NPIX_IMG] = xun[o];
  }
  // ---- lower branch: comp_phl + xuh + decomp(fhl) -> GRU(upd_l) ----
  {
    float mm[HID];
    const float* W = a.w[comp_phl_W_]; const float* g = a.w[comp_phl_g_]; const float* b = a.w[comp_phl_b_];
    #pragma unroll
    for (int o = 0; o < HID; o++) {
      float s = 0.f;
      #pragma unroll
      for (int c = 0; c < HID; c++) {
        s = fmaf(W[o * 30 + c],      vl[c],    s);
        s = fmaf(W[o * 30 + 10 + c], vp[4][c], s);
        s = fmaf(W[o * 30 + 20 + c], vp[5][c], s);
      }
      mm[o] = reluf(s * g[o] + b[o]);
    }
    float al = hq[11] + a.w[dec_fhl_ba_][0];
    const float* Wa = a.w[dec_fhl_Wa_];
    #pragma unroll
    for (int c = 0; c < HID; c++) {
      al = fmaf(Wa[256 + c], vf[c], al);
      al = fmaf(Wa[266 + c], vl[c], al);
    }
    al = sigm(al);
    out[OAT + ((size_t)NB + bimg) * NPIX_IMG + rem] = al;
    const float* Wm = a.w[dec_fhl_Wm_]; const float* gg = a.w[dec_fhl_g_]; const float* bb = a.w[dec_fhl_b_];
    #pragma unroll
    for (int o = 0; o < HID; o++) {
      float s = 0.f;
      #pragma unroll
      for (int c = 0; c < HID; c++) s = fmaf(Wm[o * 10 + c], vf[c], s);
      const float xfhl = al * reluf(s * gg[o] + bb[o]);
      mm[o] += gate * dpu[o] + (1.f - gate) * vl[o] + xfhl;   // + xuh + xfhl
    }
    float xln[HID];
    gru10(vl, mm, a.w[upd_l_Wz_], a.w[upd_l_bz_], a.w[upd_l_Wr_], a.w[upd_l_br_],
          a.w[upd_l_Wn_], a.w[upd_l_bn_], xln);
    #pragma unroll
    for (int o = 0; o < HID; o++)
      out[OXH + (size_t)NB * HID * NPIX_IMG + po + (size_t)o * NPIX_IMG] = xln[o];
  }
  // ---- per-part: satt, att_hp, dp (pdp_p), msg_hp (dec_hp) ----
  float satt[6], dp6[6][HID], mrg[6][HID];
  {
    const float* PpW = a.w[pdp_p_W_]; const float* Ppg = a.w[pdp_p_g_]; const float* Ppb = a.w[pdp_p_b_];
    const float* ApW = a.w[att_p_W_]; const float* Apb = a.w[att_p_b_];
    const float* HWa = a.w[dec_hp_Wa_]; const float* Hba = a.w[dec_hp_ba_];
    const float* HWm = a.w[dec_hp_Wm_]; const float* Hg = a.w[dec_hp_g_]; const float* Hb = a.w[dec_hp_b_];
    #pragma unroll
    for (int p = 0; p < 6; p++) {
      const bool up = (p < 4);                 // PART_PARENT = [0,0,0,0,1,1]
      float sa = Apb[p];
      float ah = pq[10 + p] + Hba[p];
      #pragma unroll
      for (int c = 0; c < HID; c++) {
        const float par = up ? vu[c] : vl[c];
        sa = fmaf(ApW[p * 10 + c], vp[p][c], sa);
        ah = fmaf(HWa[p * 276 + 256 + c], par, ah);
        ah = fmaf(HWa[p * 276 + 266 + c], vp[p][c], ah);
      }
      satt[p] = sigm(sa);
      const float atthp = sigm(ah);
      out[OHA + ((size_t)p * NB + bimg) * NPIX_IMG + rem] = 0.5f * (satt[p] + atthp);
      #pragma unroll
      for (int o = 0; o < HID; o++) {
        float sd = pq[o], sm = 0.f;
        #pragma unroll
        for (int c = 0; c < HID; c++) {
          const float par = up ? vu[c] : vl[c];
          sd = fmaf(PpW[o * 266 + 256 + c], vp[p][c], sd);
          sm = fmaf(HWm[p * 100 + o * 10 + c], par, sm);
        }
        dp6[p][o] = reluf(sd * Ppg[o] + Ppb[o]);
        mrg[p][o] = atthp * reluf(sm * Hg[p * 10 + o] + Hb[p * 10 + o]);  // msg_hp
      }
    }
  }
  // ---- edge messages + segment_max over targets; m = xpp + msg_hp ----
  {
    float xpp[6][HID];
    #pragma unroll
    for (int p = 0; p < 6; p++)
      #pragma unroll
      for (int o = 0; o < HID; o++) xpp[p][o] = -3.4e38f;
    const int ES[10] = {0, 1, 1, 1, 2, 2, 3, 4, 4, 5};
    const int ET[10] = {1, 0, 2, 4, 1, 3, 2, 1, 5, 4};
    #pragma unroll
    for (int e = 0; e < 10; e++) {
      const int s = ES[e], t = ET[e];
      const float ge = satt[s] * satt[t];
      #pragma unroll
      for (int o = 0; o < HID; o++) {
        const float v = ge * dp6[s][o] + (1.f - ge) * vp[t][o];
        xpp[t][o] = fmaxf(xpp[t][o], v);
      }
    }
    #pragma unroll
    for (int p = 0; p < 6; p++)
      #pragma unroll
      for (int o = 0; o < HID; o++) mrg[p][o] += xpp[p][o];
  }
  // ---- per-part GRU (upd_p) ----
  #pragma unroll
  for (int p = 0; p < 6; p++) {
    float xpn[HID];
    gru10(vp[p], mrg[p],
          a.w[upd_p_Wz_] + p * 200, a.w[upd_p_bz_] + p * 10,
          a.w[upd_p_Wr_] + p * 200, a.w[upd_p_br_] + p * 10,
          a.w[upd_p_Wn_] + p * 200, a.w[upd_p_bn_] + p * 10, xpn);
    #pragma unroll
    for (int o = 0; o < HID; o++)
      out[OXP + (size_t)p * NB * HID * NPIX_IMG + po + (size_t)o * NPIX_IMG] = xpn[o];
  }
}

// ---------------------------------------------------------------------------
extern "C" void kernel_launch(void* const* d_in, const int* in_sizes, int n_in,
                              void* d_out, int out_size, void* d_ws, size_t ws_size,
                              hipStream_t stream)
{
  (void)in_sizes; (void)n_in; (void)out_size; (void)ws_size;
  const float* xp   = (const float*)d_in[0];
  const float* xh   = (const float*)d_in[1];
  const float* xf   = (const float*)d_in[2];
  const float* pfea = (const float*)d_in[3];
  const float* hfea = (const float*)d_in[4];

  float* hproj = (float*)d_ws;                     // NPIX*16 floats
  float* pproj = hproj + (size_t)NPIX * 16;        // NPIX*16 floats (14.75 MB total)

  proj_wmma<<<TILES / 8, 256, 0, stream>>>(
      pfea, hfea,
      (const float*)d_in[5 + pdp_h_W_],
      (const float*)d_in[5 + dec_fhu_Wa_],
      (const float*)d_in[5 + dec_fhl_Wa_],
      (const float*)d_in[5 + pdp_p_W_],
      (const float*)d_in[5 + dec_hp_Wa_],
      hproj, pproj);

  KArgs a;
  a.xp = xp; a.xh = xh; a.xf = xf;
  a.hproj = hproj; a.pproj = pproj;
  a.out = (float*)d_out;
  for (int i = 0; i < NW_; i++) a.w[i] = (const float*)d_in[5 + i];

  fused_pw<<<NPIX / 256, 256, 0, stream>>>(a);
}